// SAGE_47991964565964
// MI455X (gfx1250) — compile-verified
//
#include <hip/hip_runtime.h>
#include <hip/hip_bf16.h>

#define N_NODES 50000
#define N_EDGES 800000
#define NFEAT   96
#define NHID    128
#define EPSN    1e-12f

typedef __attribute__((ext_vector_type(2))) float v2f;
typedef __attribute__((ext_vector_type(8))) float v8f;

// ---------------- zero workspace ----------------
__global__ void sage_zero_kernel(float* __restrict__ p, int n) {
    int i = blockIdx.x * blockDim.x + threadIdx.x;
    if (i < n) p[i] = 0.0f;
}

// ---------------- edge scatter (segment-sum + degree) ----------------
// thread = (edge, chunk-of-4-features); 24 chunks cover NFEAT=96
__global__ __launch_bounds__(256) void sage_scatter_kernel(
    const int* __restrict__ ei, const float* __restrict__ x,
    float* __restrict__ msg, float* __restrict__ deg) {
    const int CH = NFEAT / 4;                  // 24
    int t = blockIdx.x * blockDim.x + threadIdx.x;
    if (t >= N_EDGES * CH) return;
    int e = t / CH;
    int c = t - e * CH;
    int src = ei[e];
    int dst = ei[N_EDGES + e];
    const float4 v = *(const float4*)(x + (size_t)src * NFEAT + c * 4);
    float* m = msg + (size_t)dst * NFEAT + c * 4;
    unsafeAtomicAdd(m + 0, v.x);
    unsafeAtomicAdd(m + 1, v.y);
    unsafeAtomicAdd(m + 2, v.z);
    unsafeAtomicAdd(m + 3, v.w);
    if (c == 0) unsafeAtomicAdd(deg + dst, 1.0f);
}

// ---------------- fused GEMM + bias + L2-normalize ----------------
// block = 256 threads = 8 wave32s; block covers 16 nodes x 128 hidden.
// wave w owns output tile rows [blk*16, blk*16+16) x cols [w*16, w*16+16).
// D = (msg/deg) @ W_l^T + x @ W_r^T  via V_WMMA_F32_16X16X4_F32, K stepped by 4.
__global__ __launch_bounds__(256) void sage_gemm_kernel(
    const float* __restrict__ x,  const float* __restrict__ Wl,
    const float* __restrict__ bl, const float* __restrict__ Wr,
    const float* __restrict__ msg, const float* __restrict__ deg,
    float* __restrict__ out) {

    const int lane = threadIdx.x & 31;
    const int wave = threadIdx.x >> 5;     // 0..7  -> hidden-tile
    const int m16  = lane & 15;            // A row / B col within tile
    const int half = lane >> 4;            // 0/1
    const int rowBase = blockIdx.x * 16;   // 50000 = 3125*16, exact
    const int node = rowBase + m16;        // A-fragment row node
    const int col  = wave * 16 + m16;      // output column (0..127)

    const float dscale = 1.0f / fmaxf(deg[node], 1.0f);

    v8f acc = {0.f, 0.f, 0.f, 0.f, 0.f, 0.f, 0.f, 0.f};

    const float* xrow = x   + (size_t)node * NFEAT;
    const float* mrow = msg + (size_t)node * NFEAT;
    const float* wlr  = Wl  + (size_t)col  * NFEAT;   // B[k][n] = W[n][k]
    const float* wrr  = Wr  + (size_t)col  * NFEAT;

    #pragma unroll
    for (int k0 = 0; k0 < NFEAT; k0 += 4) {
        const int ka = k0 + 2 * half;      // lane's K base (K = ka + v, v in {0,1})
        float2 ax = *(const float2*)(xrow + ka);
        float2 am = *(const float2*)(mrow + ka);
        float2 wl = *(const float2*)(wlr + ka);
        float2 wr = *(const float2*)(wrr + ka);
        v2f a_g = {am.x * dscale, am.y * dscale};
        v2f a_x = {ax.x, ax.y};
        v2f b_l = {wl.x, wl.y};
        v2f b_r = {wr.x, wr.y};
        acc = __builtin_amdgcn_wmma_f32_16x16x4_f32(false, a_g, false, b_l,
                                                    (short)0, acc, false, false);
        acc = __builtin_amdgcn_wmma_f32_16x16x4_f32(false, a_x, false, b_r,
                                                    (short)0, acc, false, false);
    }

    // bias: every acc element in this lane has column `col`
    const float bv = bl[col];
    #pragma unroll
    for (int g = 0; g < 8; ++g) acc[g] += bv;

    // ---- L2 normalize across the full 128-wide row ----
    // lane holds (row = g + 8*half, col); sum squares across the 16 lanes of
    // each half-wave (the 16 columns this wave owns), then combine the 8
    // waves' partials in LDS.
    __shared__ float ssq[16];
    if (threadIdx.x < 16) ssq[threadIdx.x] = 0.0f;
    __syncthreads();

    float red[8];
    #pragma unroll
    for (int g = 0; g < 8; ++g) {
        float p = acc[g] * acc[g];
        p += __shfl_xor(p, 1, 32);
        p += __shfl_xor(p, 2, 32);
        p += __shfl_xor(p, 4, 32);
        p += __shfl_xor(p, 8, 32);
        red[g] = p;                        // per-16-lane-group sum
    }
    if (m16 == 0) {
        #pragma unroll
        for (int g = 0; g < 8; ++g)
            atomicAdd(&ssq[g + 8 * half], red[g]);   // ds_add_f32
    }
    __syncthreads();

    #pragma unroll
    for (int g = 0; g < 8; ++g) {
        const int r = g + 8 * half;
        const float inv = 1.0f / fmaxf(sqrtf(ssq[r]), EPSN);
        out[(size_t)(rowBase + r) * NHID + col] = acc[g] * inv;
    }
}

extern "C" void kernel_launch(void* const* d_in, const int* in_sizes, int n_in,
                              void* d_out, int out_size, void* d_ws, size_t ws_size,
                              hipStream_t stream) {
    const int*   ei = (const int*)d_in[0];     // [2, N_EDGES]
    const float* x  = (const float*)d_in[1];   // [N_NODES, NFEAT]
    const float* Wl = (const float*)d_in[2];   // [NHID, NFEAT]
    const float* bl = (const float*)d_in[3];   // [NHID]
    const float* Wr = (const float*)d_in[4];   // [NHID, NFEAT]
    float* out = (float*)d_out;

    float* msg = (float*)d_ws;                     // N_NODES*NFEAT
    float* deg = msg + (size_t)N_NODES * NFEAT;    // N_NODES

    int nz = N_NODES * (NFEAT + 1);
    sage_zero_kernel<<<(nz + 255) / 256, 256, 0, stream>>>((float*)d_ws, nz);

    int nsc = N_EDGES * (NFEAT / 4);
    sage_scatter_kernel<<<(nsc + 255) / 256, 256, 0, stream>>>(ei, x, msg, deg);

    sage_gemm_kernel<<<N_NODES / 16, 256, 0, stream>>>(x, Wl, bl, Wr, msg, deg, out);
}